// LogicLerpInputLayer_60215441489999
// MI455X (gfx1250) — compile-verified
//
#include <hip/hip_runtime.h>
#include <math.h>

#define N_REL  32
#define N_ENT  2048
#define N_ATTR 8
#define WIDTH  256
#define KDIM   (2 * N_REL + N_ATTR)   // 72

typedef float v2f __attribute__((ext_vector_type(2)));
typedef float v8f __attribute__((ext_vector_type(8)));

// ---------------------------------------------------------------------------
// ws layout (floats):
//   [0,            72*2048)          qsum   : rows 0..31 colsums (atomic),
//                                             rows 32..63 rowsums (direct)
//   [72*2048,      72*2048 + 72*256) sw     : softmax(weights, axis=0)
// d_out layout (floats):
//   [0,            256*2048)         outputs
//   [256*2048,     256*2048+72*2048) quantified
// ---------------------------------------------------------------------------

__global__ void zero_qsum_kernel(float* __restrict__ qsum) {
    int i = blockIdx.x * blockDim.x + threadIdx.x;
    if (i < N_REL * N_ENT) qsum[i] = 0.0f;   // only the atomic-accumulated rows
}

// One block = (relation r, 16-row tile). Reads 16*2048 floats (128 KB).
// Produces: full row sums for its 16 rows (direct store), and per-column
// partial sums flushed with global float atomics.
__global__ __launch_bounds__(256) void reduce_db_kernel(
        const float* __restrict__ db, float* __restrict__ qsum) {
    const int r        = blockIdx.x;      // 0..31
    const int tile     = blockIdx.y;      // 0..127
    const int t        = threadIdx.x;     // 0..255
    const int row_base = tile * 16;

    __shared__ float lrow[16];
    if (t < 16) lrow[t] = 0.0f;
    __syncthreads();

    const float4* mat = (const float4*)(db + (size_t)r * N_ENT * N_ENT
                                           + (size_t)row_base * N_ENT);
    float4 c0 = make_float4(0.f, 0.f, 0.f, 0.f);   // cols 4t..4t+3
    float4 c1 = make_float4(0.f, 0.f, 0.f, 0.f);   // cols 1024+4t..1024+4t+3

    #pragma unroll 4
    for (int row = 0; row < 16; ++row) {
        float4 v0 = mat[row * (N_ENT / 4) + t];
        float4 v1 = mat[row * (N_ENT / 4) + t + 256];
        c0.x += v0.x; c0.y += v0.y; c0.z += v0.z; c0.w += v0.w;
        c1.x += v1.x; c1.y += v1.y; c1.z += v1.z; c1.w += v1.w;
        float rp = (v0.x + v0.y) + (v0.z + v0.w) + (v1.x + v1.y) + (v1.z + v1.w);
        atomicAdd(&lrow[row], rp);        // ds_add_f32 (LDS float atomic)
    }
    __syncthreads();

    // axis-2 sums -> q rows [32..63]; this block owns rows fully: direct store
    if (t < 16)
        qsum[(size_t)(N_REL + r) * N_ENT + row_base + t] = lrow[t];

    // axis-1 partial colsums -> q rows [0..31]; cross-tile accumulate via atomics
    float* col = qsum + (size_t)r * N_ENT;
    const int c = 4 * t;
    unsafeAtomicAdd(&col[c + 0], c0.x);
    unsafeAtomicAdd(&col[c + 1], c0.y);
    unsafeAtomicAdd(&col[c + 2], c0.z);
    unsafeAtomicAdd(&col[c + 3], c0.w);
    unsafeAtomicAdd(&col[1024 + c + 0], c1.x);
    unsafeAtomicAdd(&col[1024 + c + 1], c1.y);
    unsafeAtomicAdd(&col[1024 + c + 2], c1.z);
    unsafeAtomicAdd(&col[1024 + c + 3], c1.w);
}

// softmax over axis 0 (72 rows) for each of the 256 columns. One block.
__global__ __launch_bounds__(WIDTH) void softmax_w_kernel(
        const float* __restrict__ w, float* __restrict__ sw) {
    const int col = threadIdx.x;
    float m = -INFINITY;
    #pragma unroll
    for (int k = 0; k < KDIM; ++k) m = fmaxf(m, w[k * WIDTH + col]);
    float s = 0.0f;
    #pragma unroll
    for (int k = 0; k < KDIM; ++k) s += __expf(w[k * WIDTH + col] - m);
    const float inv = 1.0f / s;
    #pragma unroll
    for (int k = 0; k < KDIM; ++k)
        sw[k * WIDTH + col] = __expf(w[k * WIDTH + col] - m) * inv;
}

// quantified = 1 - exp(-q); attribute rows computed directly from input.
__global__ void finalize_quant_kernel(const float* __restrict__ qsum,
                                      const float* __restrict__ attr,
                                      float* __restrict__ quant) {
    int i = blockIdx.x * blockDim.x + threadIdx.x;
    if (i >= KDIM * N_ENT) return;
    int row = i >> 11;          // /2048
    int n   = i & (N_ENT - 1);
    float v = (row < 2 * N_REL) ? qsum[i]
                                : 2.0f * attr[(row - 2 * N_REL) * N_ENT + n];
    quant[i] = 1.0f - __expf(-v);
}

// outputs[w][n] = sum_k sw[k][w] * quantified[k][n]
// M=256(w), N=2048(n), K=72. fp32 WMMA 16x16x4, one wave per 16x16 tile.
__global__ __launch_bounds__(256) void gemm_wmma_kernel(
        const float* __restrict__ sw,      // [72][256]  (A^T storage: A[m][k]=sw[k][m])
        const float* __restrict__ quant,   // [72][2048] (B[k][n])
        float* __restrict__ out) {         // [256][2048]
    const int wave = (blockIdx.x * blockDim.x + threadIdx.x) >> 5;  // 0..2047
    const int lane = threadIdx.x & 31;
    const int mt   = wave >> 7;     // 0..15
    const int nt   = wave & 127;    // 0..127
    const int half = lane >> 4;     // K-half selector per ISA 32-bit A/B layout
    const int l    = lane & 15;
    const int m    = mt * 16 + l;   // A row for this lane
    const int n    = nt * 16 + l;   // B/D column for this lane

    v8f acc = {0.f, 0.f, 0.f, 0.f, 0.f, 0.f, 0.f, 0.f};

    #pragma unroll
    for (int ks = 0; ks < KDIM / 4; ++ks) {
        const int ka = ks * 4 + half * 2;   // lanes 0-15: K=k0,k0+1 ; 16-31: K=k0+2,k0+3
        v2f A, B;
        A.x = sw[ka * WIDTH + m];
        A.y = sw[(ka + 1) * WIDTH + m];
        B.x = quant[ka * N_ENT + n];
        B.y = quant[(ka + 1) * N_ENT + n];
        acc = __builtin_amdgcn_wmma_f32_16x16x4_f32(
                /*neg_a=*/false, A, /*neg_b=*/false, B,
                /*c_mod=*/(short)0, acc, /*reuse_a=*/false, /*reuse_b=*/false);
    }

    // D layout: VGPR v -> M = v (lanes 0-15) / v+8 (lanes 16-31), N = l
    const int mrow = mt * 16 + half * 8;
    #pragma unroll
    for (int v = 0; v < 8; ++v)
        out[(size_t)(mrow + v) * N_ENT + nt * 16 + l] = acc[v];
}

extern "C" void kernel_launch(void* const* d_in, const int* in_sizes, int n_in,
                              void* d_out, int out_size, void* d_ws, size_t ws_size,
                              hipStream_t stream) {
    const float* db   = (const float*)d_in[0];   // [32,2048,2048]
    const float* attr = (const float*)d_in[1];   // [8,2048]
    const float* wts  = (const float*)d_in[2];   // [72,256]

    float* out   = (float*)d_out;                // [256,2048] ++ [72,2048]
    float* qsum  = (float*)d_ws;                 // [72,2048]
    float* sw    = qsum + KDIM * N_ENT;          // [72,256]
    float* quant = out + WIDTH * N_ENT;          // quantified slot in d_out

    zero_qsum_kernel<<<(N_REL * N_ENT + 255) / 256, 256, 0, stream>>>(qsum);
    reduce_db_kernel<<<dim3(N_REL, N_ENT / 16), 256, 0, stream>>>(db, qsum);
    softmax_w_kernel<<<1, WIDTH, 0, stream>>>(wts, sw);
    finalize_quant_kernel<<<(KDIM * N_ENT + 255) / 256, 256, 0, stream>>>(qsum, attr, quant);
    gemm_wmma_kernel<<<(16 * 128) / 8, 256, 0, stream>>>(sw, quant, out);
}